// FFT_Health_State_Analysis_36661840839136
// MI455X (gfx1250) — compile-verified
//
#include <hip/hip_runtime.h>
#include <stdint.h>

#define ROWS_DEFAULT 4096
#define COLS 16384
#define TPB 256
#define EPT (COLS / TPB)       // 64 elements per thread
#define NWAVES (TPB / 32)      // 8 waves (wave32)
#define OUTC 54

static_assert(TPB * EPT == COLS, "chunking must cover the row");
static_assert(EPT == 64, "owner lookup uses >>6");

// Monotone order-preserving map: f32 -> u32 (larger float => larger uint)
__device__ __forceinline__ uint32_t fmap(float f) {
    uint32_t b = __float_as_uint(f);
    return (b & 0x80000000u) ? ~b : (b | 0x80000000u);
}
__device__ __forceinline__ float funmap(uint32_t u) {
    uint32_t b = (u & 0x80000000u) ? (u ^ 0x80000000u) : ~u;
    return __uint_as_float(b);
}
// key = (mapped value << 32) | ~index : u64-max == (max value, tie -> min index)
__device__ __forceinline__ unsigned long long mkkey(float f, uint32_t idx) {
    return ((unsigned long long)fmap(f) << 32) | (unsigned long long)(~idx);
}

// Rescan this thread's 64-element chunk of the LDS row, return best key.
__device__ __forceinline__ unsigned long long scan_chunk(const float* row, int tid) {
    const int base = tid * EPT;
    unsigned long long best = 0ull;
    #pragma unroll 8
    for (int j = 0; j < EPT; ++j) {
        unsigned long long c = mkkey(row[base + j], (uint32_t)(base + j));
        best = (c > best) ? c : best;
    }
    return best;
}

__global__ __launch_bounds__(TPB) void topk50_rowstats_kernel(
    const float* __restrict__ in, float* __restrict__ out) {

    __shared__ float row[COLS];                       // 64 KB staged row
    __shared__ unsigned long long wkeys[NWAVES];
    __shared__ unsigned long long topKeys[50];
    __shared__ float tidxf[50];
    __shared__ float tval[50];

    const int tid  = threadIdx.x;
    const int r    = blockIdx.x;
    const int wid  = tid >> 5;
    const int lane = tid & 31;

    // ---- Stage row: CDNA5 async global -> LDS copies (ASYNCcnt path) ----
    {
        const float* gsrc = in + (size_t)r * COLS;
        // low 32 bits of a generic __shared__ pointer == LDS byte address
        const uint32_t lds_base = (uint32_t)(uintptr_t)(&row[0]);
        #pragma unroll
        for (int i = 0; i < (COLS / 4) / TPB; ++i) {   // 16 x float4 per thread
            int c = tid + TPB * i;
            uint32_t lds_addr = lds_base + (uint32_t)(c * 16);
            unsigned long long gaddr =
                (unsigned long long)(uintptr_t)(gsrc + (size_t)c * 4);
            asm volatile("global_load_async_to_lds_b128 %0, %1, off"
                         :: "v"(lds_addr), "v"(gaddr)
                         : "memory");
        }
        asm volatile("s_wait_asynccnt 0" ::: "memory");
    }
    __syncthreads();

    // ---- Per-thread best key over its chunk ----
    unsigned long long key = scan_chunk(row, tid);

    // ---- 50 rounds of block-wide argmax with incremental invalidation ----
    for (int k = 0; k < 50; ++k) {
        // intra-wave butterfly max (wave32)
        unsigned long long v = key;
        #pragma unroll
        for (int s = 16; s > 0; s >>= 1) {
            unsigned long long o = __shfl_xor(v, s, 32);
            v = (o > v) ? o : v;
        }
        if (lane == 0) wkeys[wid] = v;
        __syncthreads();

        // every thread folds the 8 wave winners (broadcast LDS reads)
        unsigned long long g = wkeys[0];
        #pragma unroll
        for (int w = 1; w < NWAVES; ++w) {
            unsigned long long o = wkeys[w];
            g = (o > g) ? o : g;
        }
        if (tid == 0) topKeys[k] = g;

        // owner invalidates the winning element and rescans its chunk
        uint32_t gidx = ~((uint32_t)g);
        if ((int)(gidx >> 6) == tid) {
            row[gidx] = -__builtin_inff();
            key = scan_chunk(row, tid);
        }
        __syncthreads();   // protect wkeys for the next round
    }

    // ---- Decode + emit the 54 per-row outputs ----
    if (tid < 50) {
        unsigned long long kk = topKeys[tid];
        tidxf[tid] = (float)(~(uint32_t)kk);
        tval[tid]  = funmap((uint32_t)(kk >> 32));
    }
    __syncthreads();

    float* orow = out + (size_t)r * OUTC;
    if (tid == 0) {
        float si = 0.0f, sv = 0.0f;
        #pragma unroll
        for (int j = 0; j < 10; ++j) {
            si += tidxf[j];
            sv += tval[j] * tval[j];
        }
        orow[0] = si * 0.1f;                 // mean of top-10 indices
        orow[1] = sqrtf(sv * 0.1f);          // rms of top-10 values
        orow[2] = tidxf[0];                  // argmax index
        orow[3] = sqrtf(tval[0] * tval[0]);  // sqrt(v0^2) == |v0|
    }
    if (tid < 50) orow[4 + tid] = tidxf[tid];
}

extern "C" void kernel_launch(void* const* d_in, const int* in_sizes, int n_in,
                              void* d_out, int out_size, void* d_ws, size_t ws_size,
                              hipStream_t stream) {
    (void)n_in; (void)d_ws; (void)ws_size; (void)out_size;
    const float* in = (const float*)d_in[0];
    float* out = (float*)d_out;
    int rows = (in_sizes && in_sizes[0] > 0) ? (in_sizes[0] / COLS) : ROWS_DEFAULT;
    if (rows <= 0) rows = ROWS_DEFAULT;
    topk50_rowstats_kernel<<<dim3(rows), dim3(TPB), 0, stream>>>(in, out);
}